// DIIR_WRAPPER_15075335209222
// MI455X (gfx1250) — compile-verified
//
#include <hip/hip_runtime.h>

typedef __attribute__((ext_vector_type(2))) float v2f;
typedef __attribute__((ext_vector_type(4))) float v4f;
typedef __attribute__((ext_vector_type(8))) float v8f;
typedef __attribute__((ext_vector_type(4))) unsigned int u32x4;
typedef __attribute__((ext_vector_type(4))) int i32x4;
typedef __attribute__((ext_vector_type(8))) int i32x8;

#define NSEC 4

// W layout (16x16 f32 matrices, row-major, 256 floats each):
//   H  @0     y1 <- u1 (Toeplitz, also y2 <- u2)
//   GK @256   y2 <- u1
//   G  @512   y1 <- s
//   GF @768   y2 <- s
//   K  @1024  s' <- u2
//   FK @1280  s' <- u1
//   F2 @1536  s' <- s   (= A^32, padded 8->16)
//   FL @1792  chunk composition matrix (= A^L, padded)

// ---------------------------------------------------------------------------
// Helpers
// ---------------------------------------------------------------------------
__device__ __forceinline__ float xswap16(float v) {
    // ds_swizzle group-of-32: and=0x1f, or=0, xor=0x10 -> half-wave swap
    return __int_as_float(__builtin_amdgcn_ds_swizzle(__float_as_int(v), 0x401f));
}

// Convert a 16x16 f32 matrix held in C/D layout (v8f) into B-operand
// fragments (4 k-slices of 4x16).
__device__ __forceinline__ void make_sfrag(v8f Sd, int hi, v2f Sf[4]) {
    float lo[8], hb[8];
    #pragma unroll
    for (int r = 0; r < 8; ++r) {
        float d  = Sd[r];
        float xo = xswap16(d);
        lo[r] = hi ? xo : d;   // S[r][n]
        hb[r] = hi ? d : xo;   // S[r+8][n]
    }
    Sf[0].x = hi ? lo[2] : lo[0];  Sf[0].y = hi ? lo[3] : lo[1];
    Sf[1].x = hi ? lo[6] : lo[4];  Sf[1].y = hi ? lo[7] : lo[5];
    Sf[2].x = hi ? hb[2] : hb[0];  Sf[2].y = hi ? hb[3] : hb[1];
    Sf[3].x = hi ? hb[6] : hb[4];  Sf[3].y = hi ? hb[7] : hb[5];
}

// A-operand fragments of a row-major 16x16 f32 matrix.
__device__ __forceinline__ void load_afrag(const float* __restrict__ M,
                                           int n, int hi, v2f f[4]) {
    #pragma unroll
    for (int kk = 0; kk < 4; ++kk) {
        const int base = n * 16 + 4 * kk + 2 * hi;
        f[kk].x = M[base];
        f[kk].y = M[base + 1];
    }
}

// ---------------------------------------------------------------------------
// Kernel 1: build all block matrices by exact simulation.
// Threads 0..15 : impulse at t=tid           -> cols of H, GK, FK
// Threads 16..31: impulse at t=tid (2nd half)-> cols of K
// Threads 32..47: unit state e_i, long sim   -> cols of G, GF, F2, FL, Gall
// ---------------------------------------------------------------------------
__global__ void __launch_bounds__(64)
diir_build_mats(const float* __restrict__ bco, const float* __restrict__ aco,
                float* __restrict__ W, float* __restrict__ gall, int lchunk) {
    const int tid  = threadIdx.x;
    const int mode = tid >> 4;          // 0,1: impulses; 2: states; 3: idle
    const int col  = tid & 15;
    if (mode >= 3) return;

    float bb0[NSEC], bb1[NSEC], bb2[NSEC], aa0[NSEC], aa1[NSEC];
    #pragma unroll
    for (int k = 0; k < NSEC; ++k) {
        bb0[k] = bco[k * 3 + 0];
        bb1[k] = bco[k * 3 + 1];
        bb2[k] = bco[k * 3 + 2];
        aa0[k] = aco[k * 2 + 0];
        aa1[k] = aco[k * 2 + 1];
    }
    float s1[NSEC] = {0.f, 0.f, 0.f, 0.f};
    float s2[NSEC] = {0.f, 0.f, 0.f, 0.f};
    if (mode == 2 && col < 8) {
        if (col & 1) s2[col >> 1] = 1.0f;
        else         s1[col >> 1] = 1.0f;
    }
    const int nsteps = (mode == 2) ? (lchunk > 32 ? lchunk : 32) : 32;
    float yr0[32];
    float c1[NSEC], c2[NSEC];   // state snapshot after 32 steps
    for (int t = 0; t < nsteps; ++t) {
        float sig = ((mode <= 1) && (t == tid)) ? 1.0f : 0.0f;
        #pragma unroll
        for (int k = 0; k < NSEC; ++k) {
            float yv = bb0[k] * sig + s1[k];
            float n1 = bb1[k] * sig - aa0[k] * yv + s2[k];
            float n2 = bb2[k] * sig - aa1[k] * yv;
            s1[k] = n1;
            s2[k] = n2;
            sig = yv;
        }
        if (t < 32) yr0[t] = sig;
        if (mode == 2 && gall != nullptr && t < lchunk)
            gall[t * 16 + col] = sig;               // row t of Gall
        if (t == 31) {
            #pragma unroll
            for (int k = 0; k < NSEC; ++k) { c1[k] = s1[k]; c2[k] = s2[k]; }
        }
    }
    float f32s[16], fLs[16];
    #pragma unroll
    for (int m = 0; m < 16; ++m) {
        f32s[m] = (m < 8) ? ((m & 1) ? c2[m >> 1] : c1[m >> 1]) : 0.0f;
        fLs[m]  = (m < 8) ? ((m & 1) ? s2[m >> 1] : s1[m >> 1]) : 0.0f;
    }

    if (mode == 0) {
        #pragma unroll
        for (int m = 0; m < 16; ++m) {
            W[   0 + m * 16 + col] = yr0[m];        // H
            W[ 256 + m * 16 + col] = yr0[16 + m];   // GK
            W[1280 + m * 16 + col] = f32s[m];       // FK
        }
    } else if (mode == 1) {
        #pragma unroll
        for (int m = 0; m < 16; ++m)
            W[1024 + m * 16 + col] = f32s[m];       // K
    } else {
        #pragma unroll
        for (int m = 0; m < 16; ++m) {
            W[ 512 + m * 16 + col] = yr0[m];        // G
            W[ 768 + m * 16 + col] = yr0[16 + m];   // GF
            W[1536 + m * 16 + col] = f32s[m];       // F2 = A^32
            W[1792 + m * 16 + col] = fLs[m];        // FL = A^lchunk
        }
    }
}

// ---------------------------------------------------------------------------
// TDM: DMA one 16(batch) x 32(time) f32 tile into LDS.
// pad_enable, interval 16 DW, amount 4 DW -> LDS row stride 40 floats with
// the t>=16 half shifted by +4 (bank-padded layout).
// ---------------------------------------------------------------------------
__device__ __forceinline__ void tdm_load_tile(unsigned lds_off,
                                              const float* gsrc,
                                              unsigned stride_elems) {
    unsigned long long ga = (unsigned long long)(size_t)gsrc;
    u32x4 g0;
    g0[0] = 1u;                                          // count=1, user desc
    g0[1] = lds_off;                                     // lds_addr (bytes)
    g0[2] = (unsigned)(ga & 0xffffffffull);              // global_addr lo
    g0[3] = (unsigned)((ga >> 32) & 0x01ffffffull)       // global_addr hi
          | (2u << 30);                                  // type = 2 (image)
    i32x8 g1;
    g1[0] = (2 << 16) | (1 << 20) | (3 << 22) | (3 << 25);
    g1[1] = 0;
    g1[2] = 0x4000;                                      // tensor_dim0 = 2^30
    g1[3] = 0x4000 | (32 << 16);                         // tensor_dim1; tile_dim0=32
    g1[4] = 16;                                          // tile_dim1=16
    g1[5] = (int)stride_elems;                           // tensor_dim0_stride
    g1[6] = 0;
    g1[7] = 0;
    i32x4 z4 = {0, 0, 0, 0};
#if __clang_major__ >= 23
    i32x8 z8 = {0, 0, 0, 0, 0, 0, 0, 0};
    __builtin_amdgcn_tensor_load_to_lds(g0, g1, z4, z4, z8, 0);
#else
    __builtin_amdgcn_tensor_load_to_lds(g0, g1, z4, z4, 0);
#endif
}

// ---------------------------------------------------------------------------
// Pass 1: per-(tile, chunk) blocked scan with ZERO initial state.
// Writes zero-state outputs to y and the chunk-final state to SZ.
// grid = (tiles, C), 1 wave per workgroup.
// ---------------------------------------------------------------------------
__global__ void __launch_bounds__(32)
diir_scan_chunks(const float* __restrict__ x, float* __restrict__ y,
                 const float* __restrict__ W, float* __restrict__ SZ,
                 int T, int L) {
    const int lane = threadIdx.x;
    const int n    = lane & 15;
    const int hi   = lane >> 4;
    const int tile = blockIdx.x;
    const int c    = blockIdx.y;
    __shared__ __align__(16) float u_lds[2][16 * 40];

    v2f Hf[4], GKf[4], Gf[4], GFf[4], Kf[4], FKf[4], F2f[4];
    load_afrag(W +    0, n, hi, Hf);
    load_afrag(W +  256, n, hi, GKf);
    load_afrag(W +  512, n, hi, Gf);
    load_afrag(W +  768, n, hi, GFf);
    load_afrag(W + 1024, n, hi, Kf);
    load_afrag(W + 1280, n, hi, FKf);
    load_afrag(W + 1536, n, hi, F2f);

    const float* xbase = x + (size_t)(tile * 16) * T + (size_t)c * L;
    float*       yrow  = y + (size_t)(tile * 16 + n) * T + (size_t)c * L;

    v8f Sd = {};  // running state, C/D layout

    tdm_load_tile((unsigned)(size_t)&u_lds[0][0], xbase, (unsigned)T);

    for (int t0 = 0; t0 < L; t0 += 32) {
        const int buf = (t0 >> 5) & 1;
        if (t0 + 32 < L) {
            asm volatile("s_wait_dscnt 0" ::: "memory");
            tdm_load_tile((unsigned)(size_t)&u_lds[buf ^ 1][0],
                          xbase + t0 + 32, (unsigned)T);
            __builtin_amdgcn_s_wait_tensorcnt(1);
        } else {
            __builtin_amdgcn_s_wait_tensorcnt(0);
        }
        asm volatile("" ::: "memory");

        const float* ub = &u_lds[buf][0];
        v2f Uf1[4], Uf2[4];
        #pragma unroll
        for (int kk = 0; kk < 4; ++kk) {
            Uf1[kk] = *(const v2f*)&ub[n * 40 + 4 * kk + 2 * hi];
            Uf2[kk] = *(const v2f*)&ub[n * 40 + 20 + 4 * kk + 2 * hi];
        }
        v2f Sf[4];
        make_sfrag(Sd, hi, Sf);

        v8f Y1 = {}, Y2 = {}, Sn = {};
        #pragma unroll
        for (int kk = 0; kk < 4; ++kk) {
            Y1 = __builtin_amdgcn_wmma_f32_16x16x4_f32(
                false, Hf[kk],  false, Uf1[kk], (short)0, Y1, false, false);
            Y2 = __builtin_amdgcn_wmma_f32_16x16x4_f32(
                false, GKf[kk], false, Uf1[kk], (short)0, Y2, false, false);
            Sn = __builtin_amdgcn_wmma_f32_16x16x4_f32(
                false, FKf[kk], false, Uf1[kk], (short)0, Sn, false, false);
        }
        #pragma unroll
        for (int kk = 0; kk < 4; ++kk) {
            Y2 = __builtin_amdgcn_wmma_f32_16x16x4_f32(
                false, Hf[kk],  false, Uf2[kk], (short)0, Y2, false, false);
            Sn = __builtin_amdgcn_wmma_f32_16x16x4_f32(
                false, Kf[kk],  false, Uf2[kk], (short)0, Sn, false, false);
        }
        #pragma unroll
        for (int kk = 0; kk < 4; ++kk) {
            Y1 = __builtin_amdgcn_wmma_f32_16x16x4_f32(
                false, Gf[kk],  false, Sf[kk], (short)0, Y1, false, false);
            Y2 = __builtin_amdgcn_wmma_f32_16x16x4_f32(
                false, GFf[kk], false, Sf[kk], (short)0, Y2, false, false);
            Sn = __builtin_amdgcn_wmma_f32_16x16x4_f32(
                false, F2f[kk], false, Sf[kk], (short)0, Sn, false, false);
        }
        Sd = Sn;

        v4f* op1 = (v4f*)(yrow + t0 + 8 * hi);
        op1[0] = (v4f){Y1[0], Y1[1], Y1[2], Y1[3]};
        op1[1] = (v4f){Y1[4], Y1[5], Y1[6], Y1[7]};
        v4f* op2 = (v4f*)(yrow + t0 + 16 + 8 * hi);
        op2[0] = (v4f){Y2[0], Y2[1], Y2[2], Y2[3]};
        op2[1] = (v4f){Y2[4], Y2[5], Y2[6], Y2[7]};
    }

    // chunk-final state, raw per-lane layout: [(tile*C + c)*256 + lane*8 + r]
    const size_t sb =
        ((size_t)blockIdx.x * gridDim.y + blockIdx.y) * 256 + (size_t)lane * 8;
    *(v4f*)&SZ[sb]     = (v4f){Sd[0], Sd[1], Sd[2], Sd[3]};
    *(v4f*)&SZ[sb + 4] = (v4f){Sd[4], Sd[5], Sd[6], Sd[7]};
}

// ---------------------------------------------------------------------------
// Pass 2: per tile, compose chunk states: S_in[c+1] = FL*S_in[c] + SZ[c].
// SZ enters directly as the WMMA C operand. grid = (tiles), 1 wave each.
// ---------------------------------------------------------------------------
__global__ void __launch_bounds__(32)
diir_chunk_scan(const float* __restrict__ SZ, float* __restrict__ SIN,
                const float* __restrict__ W, int C) {
    const int lane = threadIdx.x;
    const int n    = lane & 15;
    const int hi   = lane >> 4;
    const int tile = blockIdx.x;
    v2f FLf[4];
    load_afrag(W + 1792, n, hi, FLf);

    v8f S = {};
    for (int c = 0; c < C; ++c) {
        const size_t sb = ((size_t)tile * C + c) * 256 + (size_t)lane * 8;
        *(v4f*)&SIN[sb]     = (v4f){S[0], S[1], S[2], S[3]};
        *(v4f*)&SIN[sb + 4] = (v4f){S[4], S[5], S[6], S[7]};
        v4f z0 = *(const v4f*)&SZ[sb];
        v4f z1 = *(const v4f*)&SZ[sb + 4];
        v8f acc = {z0.x, z0.y, z0.z, z0.w, z1.x, z1.y, z1.z, z1.w};
        v2f Sf[4];
        make_sfrag(S, hi, Sf);
        #pragma unroll
        for (int kk = 0; kk < 4; ++kk)
            acc = __builtin_amdgcn_wmma_f32_16x16x4_f32(
                false, FLf[kk], false, Sf[kk], (short)0, acc, false, false);
        S = acc;
    }
}

// ---------------------------------------------------------------------------
// Pass 3: y += Gall * S_in[c] per (tile, chunk). Previously stored y loads
// land directly in the WMMA C operand. Independent waves, no serial chain.
// grid = (tiles, C), 1 wave each.
// ---------------------------------------------------------------------------
__global__ void __launch_bounds__(32)
diir_correct(float* __restrict__ y, const float* __restrict__ gall,
             const float* __restrict__ SIN, int T, int L, int C) {
    const int lane = threadIdx.x;
    const int n    = lane & 15;
    const int hi   = lane >> 4;
    const int tile = blockIdx.x;
    const int c    = blockIdx.y;

    const size_t sb = ((size_t)tile * C + c) * 256 + (size_t)lane * 8;
    v4f s0 = *(const v4f*)&SIN[sb];
    v4f s1 = *(const v4f*)&SIN[sb + 4];
    v8f S = {s0.x, s0.y, s0.z, s0.w, s1.x, s1.y, s1.z, s1.w};
    v2f Sf[4];
    make_sfrag(S, hi, Sf);

    float* yrow = y + (size_t)(tile * 16 + n) * T + (size_t)c * L;

    for (int t0 = 0; t0 < L; t0 += 32) {
        v2f Ga[4], Gb[4];
        load_afrag(gall + (size_t)t0 * 16,        n, hi, Ga);
        load_afrag(gall + (size_t)(t0 + 16) * 16, n, hi, Gb);

        v4f* op1 = (v4f*)(yrow + t0 + 8 * hi);
        v4f* op2 = (v4f*)(yrow + t0 + 16 + 8 * hi);
        v4f a0 = op1[0], a1 = op1[1];
        v4f b0 = op2[0], b1 = op2[1];
        v8f Y1 = {a0.x, a0.y, a0.z, a0.w, a1.x, a1.y, a1.z, a1.w};
        v8f Y2 = {b0.x, b0.y, b0.z, b0.w, b1.x, b1.y, b1.z, b1.w};
        #pragma unroll
        for (int kk = 0; kk < 4; ++kk) {
            Y1 = __builtin_amdgcn_wmma_f32_16x16x4_f32(
                false, Ga[kk], false, Sf[kk], (short)0, Y1, false, false);
            Y2 = __builtin_amdgcn_wmma_f32_16x16x4_f32(
                false, Gb[kk], false, Sf[kk], (short)0, Y2, false, false);
        }
        op1[0] = (v4f){Y1[0], Y1[1], Y1[2], Y1[3]};
        op1[1] = (v4f){Y1[4], Y1[5], Y1[6], Y1[7]};
        op2[0] = (v4f){Y2[0], Y2[1], Y2[2], Y2[3]};
        op2[1] = (v4f){Y2[4], Y2[5], Y2[6], Y2[7]};
    }
}

extern "C" void kernel_launch(void* const* d_in, const int* in_sizes, int n_in,
                              void* d_out, int out_size, void* d_ws, size_t ws_size,
                              hipStream_t stream) {
    const float* x = (const float*)d_in[0];  // (B, T, 1) f32
    const float* b = (const float*)d_in[1];  // (4, 3) f32
    const float* a = (const float*)d_in[2];  // (4, 2) f32
    float* yout = (float*)d_out;
    float* W    = (float*)d_ws;

    const int T = 32768;               // fixed by the reference
    const int B = in_sizes[0] / T;     // 512
    const int tiles = B / 16;          // 32
    const int C = 32;                  // chunks per row
    const int L = T / C;               // 1024 samples per chunk

    const size_t need =
        ((size_t)2048 + (size_t)L * 16 + 2 * (size_t)tiles * C * 256) * 4;

    if (ws_size >= need) {
        float* gall = W + 2048;
        float* SZ   = gall + (size_t)L * 16;
        float* SIN  = SZ + (size_t)tiles * C * 256;
        dim3 grid(tiles, C);
        diir_build_mats<<<1, 64, 0, stream>>>(b, a, W, gall, L);
        diir_scan_chunks<<<grid, 32, 0, stream>>>(x, yout, W, SZ, T, L);
        diir_chunk_scan<<<tiles, 32, 0, stream>>>(SZ, SIN, W, C);
        diir_correct<<<grid, 32, 0, stream>>>(yout, gall, SIN, T, L, C);
    } else {
        // Minimal-scratch fallback: single chunk spanning all of T
        // (zero initial state is exact); passes 2/3 unnecessary.
        float* SZ = W + 2048;
        dim3 grid(tiles, 1);
        diir_build_mats<<<1, 64, 0, stream>>>(b, a, W, nullptr, 32);
        diir_scan_chunks<<<grid, 32, 0, stream>>>(x, yout, W, SZ, T, T);
    }
}